// CGCN_28166395527614
// MI455X (gfx1250) — compile-verified
//
#include <hip/hip_runtime.h>
#include <hip/hip_bf16.h>

typedef __attribute__((ext_vector_type(2))) float v2f;
typedef __attribute__((ext_vector_type(8))) float v8f;

#define FEAT 128

// ---------------------------------------------------------------------------
// Utility kernels
// ---------------------------------------------------------------------------
__global__ __launch_bounds__(256) void k_zero(float* __restrict__ p, int count) {
    int i = blockIdx.x * 256 + threadIdx.x;
    if (i < count) p[i] = 0.0f;
}

// deg[dst] += 1 for every edge (self-loop +1 folded into rsqrt kernel)
__global__ __launch_bounds__(256) void k_degree(const int* __restrict__ dst,
                                                float* __restrict__ deg, int nedges) {
    int i = blockIdx.x * 256 + threadIdx.x;
    if (i < nedges) unsafeAtomicAdd(&deg[dst[i]], 1.0f);
}

// dis[i] = rsqrt(deg[i] + 1)   (in place over the degree buffer)
__global__ __launch_bounds__(256) void k_rsqrt(float* __restrict__ dis, int n) {
    int i = blockIdx.x * 256 + threadIdx.x;
    if (i < n) dis[i] = rsqrtf(dis[i] + 1.0f);
}

// ---------------------------------------------------------------------------
// WMMA f32 GEMM:  Out[n x 128] = A[n x 128] * W[128 x 128]
// Block = 256 threads = 8 waves; block b -> rows [16b,16b+16); wave w -> cols [16w,16w+16)
// Uses V_WMMA_F32_16X16X4_F32, K stepped by 4, 32 steps. W staged in LDS (64KB).
// ---------------------------------------------------------------------------
__global__ __launch_bounds__(256) void k_gemm_wmma(const float* __restrict__ A,
                                                   const float* __restrict__ W,
                                                   float* __restrict__ Out, int nrows) {
    __shared__ float Wlds[FEAT * FEAT];           // 64 KB
    int tid = threadIdx.x;
    // cooperative load of W into LDS (float4, fully coalesced)
    for (int i = tid * 4; i < FEAT * FEAT; i += 256 * 4) {
        *(float4*)&Wlds[i] = *(const float4*)&W[i];
    }
    __syncthreads();

    int wave  = tid >> 5;          // 0..7 -> column tile
    int lane  = tid & 31;
    int laneM = lane & 15;
    int half  = lane >> 4;         // 0 or 1
    int m0 = blockIdx.x * 16;      // nrows is a multiple of 16 (100000)
    int n0 = wave * 16;

    const float* Arow = A + (size_t)(m0 + laneM) * FEAT;
    v8f c = {};                    // f32 accumulator tile (8 VGPRs)

    #pragma unroll 4
    for (int k = 0; k < FEAT; k += 4) {
        // A 16x4 fragment: VGPR0: K = k+2*half ; VGPR1: K = k+2*half+1  (contiguous pair)
        v2f a = *(const v2f*)&Arow[k + 2 * half];
        // B 4x16 fragment from LDS: rows k+2*half, k+2*half+1; col n0+laneM
        v2f b;
        b.x = Wlds[(k + 2 * half) * FEAT + n0 + laneM];
        b.y = Wlds[(k + 2 * half + 1) * FEAT + n0 + laneM];
        // 8 args: (neg_a, A, neg_b, B, c_mod, C, reuse_a, reuse_b)
        c = __builtin_amdgcn_wmma_f32_16x16x4_f32(false, a, false, b, (short)0, c,
                                                  false, false);
    }

    // C/D layout: VGPR r -> row m0 + r + 8*half, col n0 + laneM
    float* Orow = Out + (size_t)(m0 + 8 * half) * FEAT + n0 + laneM;
    #pragma unroll
    for (int r = 0; r < 8; ++r) Orow[(size_t)r * FEAT] = c[r];
}

// ---------------------------------------------------------------------------
// agg[i][f] = h[i][f] * dis[i]^2 + bias[f]        (self-loop term + bias)
// one thread = one float4 (32 float4 per row)
// ---------------------------------------------------------------------------
__global__ __launch_bounds__(256) void k_self_init(const float* __restrict__ h,
                                                   const float* __restrict__ dis,
                                                   const float* __restrict__ bias,
                                                   float* __restrict__ agg, int nrows) {
    int i4 = blockIdx.x * 256 + threadIdx.x;
    if (i4 >= nrows * (FEAT / 4)) return;
    int row  = i4 >> 5;
    int colb = (i4 & 31) * 4;
    float d  = dis[row];
    float d2 = d * d;
    float4 hv = ((const float4*)h)[i4];
    float4 bv = *(const float4*)&bias[colb];
    float4 o;
    o.x = hv.x * d2 + bv.x;  o.y = hv.y * d2 + bv.y;
    o.z = hv.z * d2 + bv.z;  o.w = hv.w * d2 + bv.w;
    ((float4*)agg)[i4] = o;
}

// ---------------------------------------------------------------------------
// Edge scatter: one wave32 per edge; lane handles 4 features.
// agg[dst][f] += h[src][f] * dis[src]*dis[dst]
// ---------------------------------------------------------------------------
__global__ __launch_bounds__(256) void k_scatter(const float* __restrict__ h,
                                                 const float* __restrict__ dis,
                                                 const int* __restrict__ src,
                                                 const int* __restrict__ dst,
                                                 float* __restrict__ agg, int nedges) {
    int e = blockIdx.x * 8 + (threadIdx.x >> 5);
    if (e >= nedges) return;
    int lane = threadIdx.x & 31;
    int s = src[e];
    int d = dst[e];
    float cf = dis[s] * dis[d];
    float4 v = ((const float4*)(h + (size_t)s * FEAT))[lane];
    float* ap = agg + (size_t)d * FEAT + lane * 4;
    unsafeAtomicAdd(ap + 0, v.x * cf);
    unsafeAtomicAdd(ap + 1, v.y * cf);
    unsafeAtomicAdd(ap + 2, v.z * cf);
    unsafeAtomicAdd(ap + 3, v.w * cf);
}

// ---------------------------------------------------------------------------
// BatchNorm statistics: per-column sum and sum of squares.
// Block covers 128 rows; thread (col = tid&127, part = tid>>7) strides rows by 2.
// ---------------------------------------------------------------------------
__global__ __launch_bounds__(256) void k_bn_stats(const float* __restrict__ agg,
                                                  float* __restrict__ sums,
                                                  float* __restrict__ sumsq, int nrows) {
    int col  = threadIdx.x & 127;
    int part = threadIdx.x >> 7;
    int row0 = blockIdx.x * 128;
    int rend = min(row0 + 128, nrows);
    float s = 0.0f, ss = 0.0f;
    for (int r = row0 + part; r < rend; r += 2) {
        float v = agg[(size_t)r * FEAT + col];
        s += v;
        ss += v * v;
    }
    __shared__ float rs[256];
    __shared__ float rss[256];
    rs[threadIdx.x]  = s;
    rss[threadIdx.x] = ss;
    __syncthreads();
    if (part == 0) {
        unsafeAtomicAdd(&sums[col],  rs[col] + rs[col + 128]);
        unsafeAtomicAdd(&sumsq[col], rss[col] + rss[col + 128]);
    }
}

// scale/shift from batch stats: y = x*scale + shift
__global__ __launch_bounds__(128) void k_bn_finalize(const float* __restrict__ sums,
                                                     const float* __restrict__ sumsq,
                                                     const float* __restrict__ gamma,
                                                     const float* __restrict__ beta,
                                                     float* __restrict__ scale,
                                                     float* __restrict__ shift, int nrows) {
    int c = threadIdx.x;
    float inv_n = 1.0f / (float)nrows;
    float mean = sums[c] * inv_n;
    float var  = sumsq[c] * inv_n - mean * mean;
    float sc   = gamma[c] * rsqrtf(var + 1e-5f);
    scale[c] = sc;
    shift[c] = beta[c] - mean * sc;
}

// hbn = relu(agg*scale + shift)
__global__ __launch_bounds__(256) void k_bn_apply_relu(const float* __restrict__ agg,
                                                       const float* __restrict__ scale,
                                                       const float* __restrict__ shift,
                                                       float* __restrict__ hbn, int nrows) {
    int i4 = blockIdx.x * 256 + threadIdx.x;
    if (i4 >= nrows * (FEAT / 4)) return;
    int colb = (i4 & 31) * 4;
    float4 v  = ((const float4*)agg)[i4];
    float4 sc = *(const float4*)&scale[colb];
    float4 sh = *(const float4*)&shift[colb];
    float4 o;
    o.x = fmaxf(v.x * sc.x + sh.x, 0.0f);
    o.y = fmaxf(v.y * sc.y + sh.y, 0.0f);
    o.z = fmaxf(v.z * sc.z + sh.z, 0.0f);
    o.w = fmaxf(v.w * sc.w + sh.w, 0.0f);
    ((float4*)hbn)[i4] = o;
}

__global__ __launch_bounds__(256) void k_relu(float* __restrict__ p, int nrows) {
    int i4 = blockIdx.x * 256 + threadIdx.x;
    if (i4 >= nrows * (FEAT / 4)) return;
    float4 v = ((float4*)p)[i4];
    v.x = fmaxf(v.x, 0.0f); v.y = fmaxf(v.y, 0.0f);
    v.z = fmaxf(v.z, 0.0f); v.w = fmaxf(v.w, 0.0f);
    ((float4*)p)[i4] = v;
}

// ---------------------------------------------------------------------------
// Launcher
// ---------------------------------------------------------------------------
extern "C" void kernel_launch(void* const* d_in, const int* in_sizes, int n_in,
                              void* d_out, int out_size, void* d_ws, size_t ws_size,
                              hipStream_t stream) {
    const float* x     = (const float*)d_in[0];
    const float* W1    = (const float*)d_in[1];
    const float* b1    = (const float*)d_in[2];
    const float* gamma = (const float*)d_in[3];
    const float* beta  = (const float*)d_in[4];
    const float* W2    = (const float*)d_in[5];
    const float* b2    = (const float*)d_in[6];
    const int*   ei    = (const int*)d_in[7];

    int n = in_sizes[0] / FEAT;        // 100000
    int e = in_sizes[7] / 2;           // 1600000
    const int* srcIdx = ei;            // edge_index[0]
    const int* dstIdx = ei + e;        // edge_index[1]

    // workspace layout (floats): dis | sums | sumsq | scale | shift | h1 | h2
    float* ws    = (float*)d_ws;
    float* dis   = ws;                         // n   (degree accum, then rsqrt in place)
    float* sums  = dis + n;                    // 128
    float* sumsq = sums + FEAT;                // 128
    float* scale = sumsq + FEAT;               // 128
    float* shift = scale + FEAT;               // 128
    float* h1    = shift + FEAT;               // n*128
    float* h2    = h1 + (size_t)n * FEAT;      // n*128
    float* out   = (float*)d_out;

    int q  = n * (FEAT / 4);                   // float4 count of a feature matrix
    int zb = (n + 512 + 255) / 256;            // zero dis + sums + sumsq (contiguous)
    int eb = (e + 255) / 256;
    int qb = (q + 255) / 256;
    int sb = (e + 7) / 8;                      // scatter: 8 edges (waves) per block
    int gb = n / 16;                           // GEMM row tiles (n % 16 == 0)

    k_zero<<<zb, 256, 0, stream>>>(ws, n + 512);
    k_degree<<<eb, 256, 0, stream>>>(dstIdx, dis, e);
    k_rsqrt<<<(n + 255) / 256, 256, 0, stream>>>(dis, n);

    // ---- layer 1 ----
    k_gemm_wmma<<<gb, 256, 0, stream>>>(x, W1, h1, n);
    k_self_init<<<qb, 256, 0, stream>>>(h1, dis, b1, h2, n);
    k_scatter<<<sb, 256, 0, stream>>>(h1, dis, srcIdx, dstIdx, h2, e);

    // ---- batchnorm + relu ----
    k_bn_stats<<<(n + 127) / 128, 256, 0, stream>>>(h2, sums, sumsq, n);
    k_bn_finalize<<<1, 128, 0, stream>>>(sums, sumsq, gamma, beta, scale, shift, n);
    k_bn_apply_relu<<<qb, 256, 0, stream>>>(h2, scale, shift, h1, n);

    // ---- layer 2 ----
    k_gemm_wmma<<<gb, 256, 0, stream>>>(h1, W2, h2, n);
    k_self_init<<<qb, 256, 0, stream>>>(h2, dis, b2, out, n);
    k_scatter<<<sb, 256, 0, stream>>>(h2, dis, srcIdx, dstIdx, out, e);
    k_relu<<<qb, 256, 0, stream>>>(out, n);
}